// DWT_2D_19164144075605
// MI455X (gfx1250) — compile-verified
//
#include <hip/hip_runtime.h>
#include <hip/hip_bf16.h>
#include <stdint.h>

// ---------------------------------------------------------------------------
// Haar 2-D DWT, (16,64,256,256) f32 -> 4 x (16,64,128,128) f32 (ll,lh,hl,hh).
// Pure streaming problem: 512 MiB total traffic, ~23 us floor @ 23.3 TB/s,
// ~0.26 flop/byte -> WMMA is deliberately NOT used (it would add ~85x flops
// and layout traffic to an HBM-bound kernel).
// CDNA5 path used instead: async global->LDS DMA (ASYNCcnt) with a 3-deep
// software pipeline in GVS addressing mode + non-temporal cache hints,
// conflict-free ds_load_b128 consumption, coalesced non-temporal b64 stores.
// ---------------------------------------------------------------------------

typedef __attribute__((ext_vector_type(2))) float f2;
typedef __attribute__((ext_vector_type(4))) float f4;

#define DWT_TPB 256
#define DWT_W 256            // input row length (floats)
#define DWT_STAGE_ROWS 8     // input rows per pipeline stage
#define DWT_STAGE_FLOATS (DWT_STAGE_ROWS * DWT_W)   // 2048 floats = 8 KB
#define DWT_STAGE_BYTES  (DWT_STAGE_FLOATS * 4)     // 8192
#define DWT_NBUF 4           // circular LDS buffers (32 KB total)
#define DWT_NS 16            // stages per block (16*8 = 128 input rows)
#define DWT_DEPTH 3          // async prefetch distance (stages)

// gfx1250 async DMA, GVS mode: copies 16 bytes from global sbase plus voff to LDS at lds_byte_addr,
// tracked by ASYNCcnt. Non-temporal: input is streamed exactly once.
__device__ __forceinline__ void async_ld16(unsigned lds_byte_addr,
                                           unsigned voff,
                                           unsigned long long sbase) {
  asm volatile("global_load_async_to_lds_b128 %0, %1, %2 th:TH_LOAD_NT"
               :
               : "v"(lds_byte_addr), "v"(voff), "s"(sbase)
               : "memory");
}

#define WAIT_ASYNC(n) asm volatile("s_wait_asynccnt " #n ::: "memory")

__global__ __launch_bounds__(DWT_TPB)
void haar_dwt2d_kernel(const float* __restrict__ x, float* __restrict__ out) {
  __shared__ __align__(16) float smem[DWT_NBUF * DWT_STAGE_FLOATS]; // 32 KB

  const int tid   = threadIdx.x;
  const int blk   = blockIdx.x;       // 2048 blocks
  const int plane = blk >> 1;         // 0..1023  (B*C)
  const int half  = blk & 1;          // top/bottom 128 input rows

  // Block-uniform 64-bit base of this block's 128 contiguous input rows.
  const unsigned long long sbase =
      (unsigned long long)(uintptr_t)(x + ((size_t)plane * DWT_W +
                                           (size_t)half * 128) * DWT_W);

  // Low 32 bits of a flat shared-aperture address == LDS byte offset (gfx1250).
  const unsigned lds0  = (unsigned)(uintptr_t)(&smem[0]);
  const unsigned vbase = (unsigned)tid * 16u;   // per-lane 16B slice

  // ---- prime the pipeline: issue DEPTH stages of async DMA ------------------
#pragma unroll
  for (int s = 0; s < DWT_DEPTH; ++s) {
    const unsigned goff = (unsigned)s * DWT_STAGE_BYTES + vbase;
    const unsigned d    = lds0 + (unsigned)((s % DWT_NBUF) * DWT_STAGE_BYTES) + vbase;
    async_ld16(d,         goff,         sbase);
    async_ld16(d + 4096u, goff + 4096u, sbase);
  }

  const int orow = tid >> 6;   // 0..3 : output row within stage
  const int cp   = tid & 63;   // 0..63: output column pair (cols 2cp, 2cp+1)
  const float c = 0.70710678118654752440f;  // 1/sqrt(2)

  const size_t SB = (size_t)1024 * 128 * 128;  // sub-band stride in d_out
  const size_t obase = (size_t)plane * (128 * 128)
                     + (size_t)(half * 64 + orow) * 128
                     + (size_t)(2 * cp);

  for (int s = 0; s < DWT_NS; ++s) {
    // ---- issue async DMA for stage s+DEPTH (its buffer was freed by the
    //      trailing barrier of iteration s-1)
    const int ahead = s + DWT_DEPTH;
    if (ahead < DWT_NS) {
      const unsigned goff = (unsigned)ahead * DWT_STAGE_BYTES + vbase;
      const unsigned d    = lds0 + (unsigned)((ahead % DWT_NBUF) * DWT_STAGE_BYTES) + vbase;
      async_ld16(d,         goff,         sbase);
      async_ld16(d + 4096u, goff + 4096u, sbase);
    }

    // ---- wait until this wave's slice of stage s landed. Async loads retire
    // in order; after the wait, only the 2 asyncs of each newer in-flight
    // stage remain outstanding.
    const int rem = DWT_NS - 1 - s;
    if (rem >= 3)      { WAIT_ASYNC(6); }
    else if (rem == 2) { WAIT_ASYNC(4); }
    else if (rem == 1) { WAIT_ASYNC(2); }
    else               { WAIT_ASYNC(0); }
    __syncthreads();   // every wave waited -> whole stage visible in LDS

    // ---- consume stage s: vertical + horizontal Haar butterfly -------------
    const float* buf = smem + (s % DWT_NBUF) * DWT_STAGE_FLOATS;
    const f4 t = *(const f4*)(buf + (2 * orow)     * DWT_W + cp * 4);
    const f4 b = *(const f4*)(buf + (2 * orow + 1) * DWT_W + cp * 4);

    // row (vertical) pass: combine input rows 2i and 2i+1
    const float lo0 = (t.x + b.x) * c, lo1 = (t.y + b.y) * c;
    const float lo2 = (t.z + b.z) * c, lo3 = (t.w + b.w) * c;
    const float hi0 = (t.x - b.x) * c, hi1 = (t.y - b.y) * c;
    const float hi2 = (t.z - b.z) * c, hi3 = (t.w - b.w) * c;

    // column (horizontal) pass -> two output columns per sub-band
    f2 ll; ll.x = (lo0 + lo1) * c; ll.y = (lo2 + lo3) * c;
    f2 lh; lh.x = (lo0 - lo1) * c; lh.y = (lo2 - lo3) * c;
    f2 hl; hl.x = (hi0 + hi1) * c; hl.y = (hi2 + hi3) * c;
    f2 hh; hh.x = (hi0 - hi1) * c; hh.y = (hi2 - hi3) * c;

    const size_t o = obase + (size_t)(s * 4) * 128;
    __builtin_nontemporal_store(ll, (f2*)(out + o));
    __builtin_nontemporal_store(lh, (f2*)(out + SB + o));
    __builtin_nontemporal_store(hl, (f2*)(out + 2 * SB + o));
    __builtin_nontemporal_store(hh, (f2*)(out + 3 * SB + o));

    __syncthreads();   // all reads of buf done before it is re-targeted
  }
}

extern "C" void kernel_launch(void* const* d_in, const int* in_sizes, int n_in,
                              void* d_out, int out_size, void* d_ws, size_t ws_size,
                              hipStream_t stream) {
  (void)in_sizes; (void)n_in; (void)out_size; (void)d_ws; (void)ws_size;
  const float* x = (const float*)d_in[0];
  float* out = (float*)d_out;
  // 1024 planes * 2 half-planes per block, 8 waves/block (wave32)
  haar_dwt2d_kernel<<<dim3(2048), dim3(DWT_TPB), 0, stream>>>(x, out);
}